// domain_img_cls_GRAM_14267881357656
// MI455X (gfx1250) — compile-verified
//
#include <hip/hip_runtime.h>
#include <math.h>

#define DIM 192
#define HW 1024
#define GRAM_DIM 18528
#define H1_DIM 9264
#define H2_DIM 4632
#define H3_DIM 64
#define BATCH 32

typedef __attribute__((ext_vector_type(2))) float v2f;
typedef __attribute__((ext_vector_type(8))) float v8f;

__device__ __forceinline__ v8f wmma_f32(v2f a, v2f b, v8f c) {
  // (neg_a, A, neg_b, B, c_mod, C, reuse_a, reuse_b) -- f32 16x16x4
  return __builtin_amdgcn_wmma_f32_16x16x4_f32(false, a, false, b, (short)0, c,
                                               false, false);
}

// Async DMA: global -> LDS, 16 bytes per lane, tracked by ASYNCcnt.
// Generic pointers to LDS carry the LDS byte offset in their low 32 bits
// (ISA 10.2).
__device__ __forceinline__ void async_load_f4(const float* gptr, void* lptr) {
  unsigned lds = (unsigned)(unsigned long long)lptr;
  asm volatile("global_load_async_to_lds_b128 %0, %1, off"
               :: "v"(lds), "v"(gptr) : "memory");
}
__device__ __forceinline__ void wait_async0() {
  asm volatile("s_wait_asynccnt 0x0" ::: "memory");
}

// ---------------------------------------------------------------------------
// Kernel 1: per-batch Gram upper-triangle via V_WMMA_F32_16X16X4_F32.
// grid = (78 upper-tri tiles, 32 batches), block = 1 wave. x is L2-resident
// (25 MB); many independent 1-wave blocks hide the stage latency.
// ---------------------------------------------------------------------------
#define GK_CHUNK 64
#define GSTRIDE 68

__global__ void gram_kernel(const float* __restrict__ x, float* __restrict__ g) {
  __shared__ float As[16 * GSTRIDE];
  __shared__ float Bs[16 * GSTRIDE];
  const int t = threadIdx.x;        // 0..31
  const int b = blockIdx.y;
  int i = 0, rem = blockIdx.x;      // row-major upper-tri tile -> (i, j), j>=i
  while (rem >= (12 - i)) { rem -= (12 - i); ++i; }
  const int j = i + rem;

  const float* feat = x + (size_t)b * DIM * HW;
  const float* arow = feat + (size_t)(i * 16) * HW;
  const float* brow = feat + (size_t)(j * 16) * HW;

  const int lr = t & 15;
  const int half = t >> 4;

  v8f c = {};
  for (int kc = 0; kc < HW; kc += GK_CHUNK) {
    const int fi = t & 15;
    const int r0 = t >> 4;
#pragma unroll
    for (int it = 0; it < 8; ++it) {
      const int r = r0 + it * 2;
      float4 av = *(const float4*)(arow + (size_t)r * HW + kc + fi * 4);
      float4 bv = *(const float4*)(brow + (size_t)r * HW + kc + fi * 4);
      *(float4*)&As[r * GSTRIDE + fi * 4] = av;
      *(float4*)&Bs[r * GSTRIDE + fi * 4] = bv;
    }
    __syncthreads();
#pragma unroll
    for (int kk = 0; kk < GK_CHUNK; kk += 4) {
      const int cb = kk + 2 * half;     // lanes 16-31 hold K+2,K+3
      v2f a  = *(const v2f*)&As[lr * GSTRIDE + cb];
      v2f bb = *(const v2f*)&Bs[lr * GSTRIDE + cb];
      c = wmma_f32(a, bb, c);
    }
    __syncthreads();
  }

#pragma unroll
  for (int v = 0; v < 8; ++v) {
    const int m  = v + 8 * half;
    const int r  = i * 16 + m;
    const int cc = j * 16 + lr;
    if (cc >= r) {
      const int idx = r * DIM - (r * (r - 1)) / 2 + (cc - r);
      g[(size_t)b * GRAM_DIM + idx] = c[v];
    }
  }
}

// ---------------------------------------------------------------------------
// Kernel 2: out[32,N] = leaky_relu(A[32,K] @ W[N,K]^T + bias), W row-major.
// 256 threads / 8 waves; 128 output columns per block, each wave owns a
// 16-column tile and both 16-row M tiles (W streamed exactly once chip-wide).
// Double-buffered GLOBAL_LOAD_ASYNC_TO_LDS_B128 pipeline: the DMA for chunk
// ch+1 overlaps the 32 WMMAs on chunk ch, keeping the 23.3 TB/s HBM path
// saturated. N-tail rows clamped (outputs masked); K-tail zero-filled.
// ---------------------------------------------------------------------------
#define FK_CHUNK 64
#define WSTR 68
#define NBLK 128

__global__ void fc_kernel(const float* __restrict__ A, const float* __restrict__ W,
                          const float* __restrict__ bias, float* __restrict__ out,
                          int N, int K) {
  __shared__ float Ws[2 * NBLK * WSTR];   // 69.6 KB
  __shared__ float Az[2 * BATCH * WSTR];  // 17.4 KB
  const int t = threadIdx.x;      // 0..255
  const int wave = t >> 5;
  const int lane = t & 31;
  const int lr = lane & 15;
  const int half = lane >> 4;
  const int n0 = blockIdx.x * NBLK;

  const int fi = t & 15;          // float4 column within 64-float chunk row
  const int r0 = t >> 4;          // 0..15

  auto stage = [&](int ch, int buf) {
    const int k = ch * FK_CHUNK + fi * 4;
    float* WsB = &Ws[buf * NBLK * WSTR];
    float* AzB = &Az[buf * BATCH * WSTR];
#pragma unroll
    for (int it = 0; it < 8; ++it) {        // 128 W rows
      const int r = r0 + it * 16;
      float* dst = &WsB[r * WSTR + fi * 4];
      if (k < K) {
        int n = n0 + r;
        if (n >= N) n = N - 1;              // clamp: columns >= N never stored
        async_load_f4(W + (size_t)n * K + k, dst);
      } else {
        *(float4*)dst = make_float4(0.f, 0.f, 0.f, 0.f);
      }
    }
#pragma unroll
    for (int it = 0; it < 2; ++it) {        // 32 A rows
      const int r = r0 + it * 16;
      float* dst = &AzB[r * WSTR + fi * 4];
      if (k < K) async_load_f4(A + (size_t)r * K + k, dst);
      else       *(float4*)dst = make_float4(0.f, 0.f, 0.f, 0.f);
    }
  };

  v8f c0 = {};
  v8f c1 = {};
  const int nch = (K + FK_CHUNK - 1) / FK_CHUNK;

  stage(0, 0);
  for (int ch = 0; ch < nch; ++ch) {
    const int p = ch & 1;
    wait_async0();                 // my chunk-ch DMAs have landed in LDS
    __syncthreads();               // everyone's landed; other buffer is free
    if (ch + 1 < nch) stage(ch + 1, 1 - p);   // DMA overlaps compute below

    const float* WsB = &Ws[p * NBLK * WSTR];
    const float* AzB = &Az[p * BATCH * WSTR];
#pragma unroll
    for (int kk = 0; kk < FK_CHUNK; kk += 4) {
      const int cb = kk + 2 * half;
      v2f bfrag = *(const v2f*)&WsB[(16 * wave + lr) * WSTR + cb];
      v2f a0 = *(const v2f*)&AzB[lr * WSTR + cb];
      v2f a1 = *(const v2f*)&AzB[(lr + 16) * WSTR + cb];
      c0 = wmma_f32(a0, bfrag, c0);
      c1 = wmma_f32(a1, bfrag, c1);
    }
  }

  const int n = n0 + 16 * wave + lr;
  if (n < N) {
    const float bv = bias[n];
#pragma unroll
    for (int v = 0; v < 8; ++v) {
      const int m0 = v + 8 * half;
      float x0 = c0[v] + bv;
      x0 = x0 > 0.f ? x0 : 0.01f * x0;
      out[(size_t)m0 * N + n] = x0;
      float x1 = c1[v] + bv;
      x1 = x1 > 0.f ? x1 : 0.01f * x1;
      out[(size_t)(m0 + 16) * N + n] = x1;
    }
  }
}

// ---------------------------------------------------------------------------
// Kernel 3: l2-normalize + sigmoid head (dot(h3,W4)/max(||h3||,eps)).
// ---------------------------------------------------------------------------
__global__ void head_kernel(const float* __restrict__ h3, const float* __restrict__ W4,
                            const float* __restrict__ b4, float* __restrict__ out) {
  const int b = threadIdx.x;        // 32 threads
  float ss = 0.f, dot = 0.f;
  for (int jj = 0; jj < H3_DIM; ++jj) {
    const float v = h3[b * H3_DIM + jj];
    ss += v * v;
    dot += v * W4[jj];
  }
  const float nrm = fmaxf(sqrtf(ss), 1e-12f);
  const float z = dot / nrm + b4[0];
  out[b] = 1.f / (1.f + __expf(-z));
}

extern "C" void kernel_launch(void* const* d_in, const int* in_sizes, int n_in,
                              void* d_out, int out_size, void* d_ws, size_t ws_size,
                              hipStream_t stream) {
  const float* x  = (const float*)d_in[0];
  const float* W1 = (const float*)d_in[1];
  const float* b1 = (const float*)d_in[2];
  const float* W2 = (const float*)d_in[3];
  const float* b2 = (const float*)d_in[4];
  const float* W3 = (const float*)d_in[5];
  const float* b3 = (const float*)d_in[6];
  const float* W4 = (const float*)d_in[7];
  const float* b4 = (const float*)d_in[8];
  float* out = (float*)d_out;

  float* g  = (float*)d_ws;                         // [32, 18528]
  float* h1 = g  + (size_t)BATCH * GRAM_DIM;        // [32, 9264]
  float* h2 = h1 + (size_t)BATCH * H1_DIM;          // [32, 4632]
  float* h3 = h2 + (size_t)BATCH * H2_DIM;          // [32, 64]

  gram_kernel<<<dim3(78, BATCH), 32, 0, stream>>>(x, g);
  fc_kernel<<<(H1_DIM + NBLK - 1) / NBLK, 256, 0, stream>>>(g,  W1, b1, h1, H1_DIM, GRAM_DIM);
  fc_kernel<<<(H2_DIM + NBLK - 1) / NBLK, 256, 0, stream>>>(h1, W2, b2, h2, H2_DIM, H1_DIM);
  fc_kernel<<<1, 256, 0, stream>>>(h2, W3, b3, h3, H3_DIM, H2_DIM);
  head_kernel<<<1, 32, 0, stream>>>(h3, W4, b4, out);
}